// Multi_Self_Attention_Head_Block_51067161149562
// MI455X (gfx1250) — compile-verified
//
#include <hip/hip_runtime.h>
#include <hip/hip_bf16.h>

// Multi-head self-attention (B=2,S=2048,D=64,H=12), fp32 throughout,
// all GEMM stages on v_wmma_f32_16x16x4_f32 (native fp32 WMMA, wave32).

#define B_  2
#define S_  2048
#define D_  64
#define H_  12
#define HD  768           // H_*D_
#define NEGV (-99999.0f)

typedef __attribute__((ext_vector_type(2))) float v2f;
typedef __attribute__((ext_vector_type(8))) float v8f;

// D = A(16x4 f32) * B(4x16 f32) + C(16x16 f32)
__device__ __forceinline__ v8f wmma4(v2f a, v2f b, v8f c) {
    return __builtin_amdgcn_wmma_f32_16x16x4_f32(
        /*neg_a=*/false, a, /*neg_b=*/false, b,
        /*c_mod=*/(short)0, c, /*reuse_a=*/false, /*reuse_b=*/false);
}

// ---------------------------------------------------------------------------
// Kernel 1: QKV projection. x[4096x64] @ W[64x768] + b -> [B,H,S,64] scratch.
// One wave computes one 16x16 output tile; K-loop of 16 WMMA steps (k=4 each).
// ---------------------------------------------------------------------------
__global__ __launch_bounds__(32)
void qkv_proj_kernel(const float* __restrict__ x,
                     const float* __restrict__ Wq, const float* __restrict__ bq,
                     const float* __restrict__ Wk, const float* __restrict__ bk,
                     const float* __restrict__ Wv, const float* __restrict__ bv,
                     float* __restrict__ ws) {
    const int lane = threadIdx.x;
    const int lo = lane & 15, hi = lane >> 4;           // hi in {0,1}
    const int n0 = blockIdx.x * 16;                     // 0..767 col tile
    const int m0 = blockIdx.y * 16;                     // 0..4095 row tile
    const int which = blockIdx.z;                       // 0=Q 1=K 2=V

    const float* W    = (which == 0) ? Wq : (which == 1) ? Wk : Wv;
    const float* bias = (which == 0) ? bq : (which == 1) ? bk : bv;
    float* out = ws + (size_t)which * (size_t)(B_ * S_ * HD);

    const float* xrow = x + (size_t)(m0 + lo) * D_;     // A-matrix row for this lane
    v8f acc = {};
    #pragma unroll
    for (int c = 0; c < 16; ++c) {
        const int kk = 4 * c + 2 * hi;
        v2f a = *(const v2f*)(xrow + kk);               // A(m=lo, k=kk..kk+1)
        v2f b;                                          // B(k=kk..kk+1, n=lo)
        b.x = W[(size_t)kk       * HD + n0 + lo];
        b.y = W[(size_t)(kk + 1) * HD + n0 + lo];
        acc = wmma4(a, b, acc);
    }
    const float bn = bias[n0 + lo];
    const int h    = n0 / 64;                           // tile never crosses heads
    const int dcol = (n0 & 63) + lo;
    #pragma unroll
    for (int v = 0; v < 8; ++v) {
        const int m = m0 + v + 8 * hi;                  // C/D layout row
        const int bb = m / S_, s = m % S_;
        out[(((size_t)bb * H_ + h) * S_ + s) * D_ + dcol] = acc[v] + bn;
    }
}

// ---------------------------------------------------------------------------
// Kernel 2: flash-style causal attention with the quirky mask.
// One wave owns a 16-row q tile of one (b,h); streams k tiles 0..q0.
// ---------------------------------------------------------------------------
__global__ __launch_bounds__(32)
void attn_kernel(const float* __restrict__ ws_q, const float* __restrict__ ws_k,
                 const float* __restrict__ ws_v, float* __restrict__ ws_o) {
    __shared__ float pl[16 * 17];                       // padded P bounce tile
    const int lane = threadIdx.x;
    const int lo = lane & 15, hi = lane >> 4;
    const int bh = blockIdx.x >> 7;                     // 0..23
    const int q0 = (blockIdx.x & 127) * 16;

    const size_t base = (size_t)bh * S_ * D_;
    const float* Q = ws_q + base;
    const float* K = ws_k + base;
    const float* V = ws_v + base;

    // Preload the Q tile in A-layout: 16 chunks of v2f.
    v2f qa[16];
    {
        const float* qrow = Q + (size_t)(q0 + lo) * D_;
        #pragma unroll
        for (int c = 0; c < 16; ++c) qa[c] = *(const v2f*)(qrow + 4 * c + 2 * hi);
    }

    v8f acc0 = {}, acc1 = {}, acc2 = {}, acc3 = {};     // O tile, 16x64
    float rm[8], rl[8];                                 // per-row max / sum (C/D layout)
    #pragma unroll
    for (int v = 0; v < 8; ++v) { rm[v] = -3.0e38f; rl[v] = 0.0f; }

    for (int k0 = 0; k0 <= q0; k0 += 16) {
        // ---- S = Q * K^T (16 WMMA steps over d) ----
        v8f s = {};
        const float* krow = K + (size_t)(k0 + lo) * D_;
        #pragma unroll
        for (int c = 0; c < 16; ++c) {
            v2f kb = *(const v2f*)(krow + 4 * c + 2 * hi);
            s = wmma4(qa[c], kb, s);
        }
        // ---- quirky causal mask + online softmax ----
        const int kcol = k0 + lo;
        float p[8];
        #pragma unroll
        for (int v = 0; v < 8; ++v) {
            const int qr = q0 + v + 8 * hi;
            float val = s[v];
            if (kcol > qr || val == 0.0f) val = NEGV;   // tril + exact-zero quirk
            p[v] = val;
        }
        #pragma unroll
        for (int v = 0; v < 8; ++v) {
            float t = p[v];                             // row-max within 16-lane half
            t = fmaxf(t, __shfl_xor(t, 1));
            t = fmaxf(t, __shfl_xor(t, 2));
            t = fmaxf(t, __shfl_xor(t, 4));
            t = fmaxf(t, __shfl_xor(t, 8));
            const float nm = fmaxf(rm[v], t);
            const float sc = __expf(rm[v] - nm);
            rm[v] = nm;
            const float e = __expf(p[v] - nm);
            p[v] = e;
            float ts = e;                               // row-sum of exp
            ts += __shfl_xor(ts, 1);
            ts += __shfl_xor(ts, 2);
            ts += __shfl_xor(ts, 4);
            ts += __shfl_xor(ts, 8);
            rl[v] = rl[v] * sc + ts;
            acc0[v] *= sc; acc1[v] *= sc; acc2[v] *= sc; acc3[v] *= sc;
        }
        // ---- C/D-layout P -> LDS -> A-layout chunks ----
        #pragma unroll
        for (int v = 0; v < 8; ++v) pl[(v + 8 * hi) * 17 + lo] = p[v];
        __syncthreads();                                // single wave: DS-cnt fence
        v2f pa[4];
        #pragma unroll
        for (int c = 0; c < 4; ++c) {
            pa[c].x = pl[lo * 17 + 4 * c + 2 * hi];
            pa[c].y = pl[lo * 17 + 4 * c + 2 * hi + 1];
        }
        __syncthreads();
        // ---- O += P * V (4 chunks x 4 n-tiles = 16 WMMA) ----
        #pragma unroll
        for (int c = 0; c < 4; ++c) {
            const int r = k0 + 4 * c + 2 * hi;
            const float* v0 = V + (size_t)r * D_;
            const float* v1 = V + (size_t)(r + 1) * D_;
            v2f b0; b0.x = v0[lo];      b0.y = v1[lo];      acc0 = wmma4(pa[c], b0, acc0);
            v2f b1; b1.x = v0[16 + lo]; b1.y = v1[16 + lo]; acc1 = wmma4(pa[c], b1, acc1);
            v2f b2; b2.x = v0[32 + lo]; b2.y = v1[32 + lo]; acc2 = wmma4(pa[c], b2, acc2);
            v2f b3; b3.x = v0[48 + lo]; b3.y = v1[48 + lo]; acc3 = wmma4(pa[c], b3, acc3);
        }
    }

    // ---- normalize + write O as [B,S,H*D] ----
    const int bb = bh / H_, h = bh % H_;
    #pragma unroll
    for (int v = 0; v < 8; ++v) {
        const float inv = 1.0f / rl[v];
        const int srow = q0 + v + 8 * hi;
        const size_t o = ((size_t)bb * S_ + srow) * HD + h * 64 + lo;
        ws_o[o]      = acc0[v] * inv;
        ws_o[o + 16] = acc1[v] * inv;
        ws_o[o + 32] = acc2[v] * inv;
        ws_o[o + 48] = acc3[v] * inv;
    }
}

// ---------------------------------------------------------------------------
// Kernel 3: output projection. O[4096x768] @ Wo[768x64] + bo -> out[4096x64].
// ---------------------------------------------------------------------------
__global__ __launch_bounds__(32)
void out_proj_kernel(const float* __restrict__ o_ws, const float* __restrict__ Wo,
                     const float* __restrict__ bo, float* __restrict__ out) {
    const int lane = threadIdx.x;
    const int lo = lane & 15, hi = lane >> 4;
    const int n0 = blockIdx.x * 16;                     // 0..63
    const int m0 = blockIdx.y * 16;                     // 0..4095
    const float* arow = o_ws + (size_t)(m0 + lo) * HD;
    v8f acc = {};
    for (int c = 0; c < 192; ++c) {
        const int kk = 4 * c + 2 * hi;
        v2f a = *(const v2f*)(arow + kk);
        v2f b;
        b.x = Wo[(size_t)kk       * D_ + n0 + lo];
        b.y = Wo[(size_t)(kk + 1) * D_ + n0 + lo];
        acc = wmma4(a, b, acc);
    }
    const float bn = bo[n0 + lo];
    #pragma unroll
    for (int v = 0; v < 8; ++v)
        out[(size_t)(m0 + v + 8 * hi) * D_ + n0 + lo] = acc[v] + bn;
}

extern "C" void kernel_launch(void* const* d_in, const int* in_sizes, int n_in,
                              void* d_out, int out_size, void* d_ws, size_t ws_size,
                              hipStream_t stream) {
    const float* x  = (const float*)d_in[0];
    const float* Wq = (const float*)d_in[1];
    const float* bq = (const float*)d_in[2];
    const float* Wk = (const float*)d_in[3];
    const float* bk = (const float*)d_in[4];
    const float* Wv = (const float*)d_in[5];
    const float* bv = (const float*)d_in[6];
    const float* Wo = (const float*)d_in[7];
    const float* bo = (const float*)d_in[8];
    float* out = (float*)d_out;

    float* ws = (float*)d_ws;
    const size_t QKV = (size_t)B_ * S_ * HD;            // 3,145,728 floats each
    float* q = ws;
    float* k = ws + QKV;
    float* v = ws + 2 * QKV;
    float* o = ws + 3 * QKV;                            // total 48 MB of d_ws

    // 1) QKV projections: 48 n-tiles x 256 m-tiles x 3 matrices, 1 wave/tile.
    qkv_proj_kernel<<<dim3(HD / 16, (B_ * S_) / 16, 3), 32, 0, stream>>>(
        x, Wq, bq, Wk, bk, Wv, bv, ws);

    // 2) Attention: 24 (b,h) x 128 q-tiles, one wave each.
    attn_kernel<<<dim3(B_ * H_ * (S_ / 16)), 32, 0, stream>>>(q, k, v, o);

    // 3) Output projection: 4 n-tiles x 256 m-tiles.
    out_proj_kernel<<<dim3(D_ / 16, (B_ * S_) / 16), 32, 0, stream>>>(o, Wo, bo, out);
}